// FinalLayer_54881092108533
// MI455X (gfx1250) — compile-verified
//
#include <hip/hip_runtime.h>

// Problem sizes (fixed by the reference)
#define NB      4
#define NGRID   512
#define NTARGET 1024
#define NBASIS  5
#define NCH     8

#define GCHUNK    128            // g-loop tile staged in LDS
#define HS_STRIDE 133            // padded row stride (133%64=5 -> conflict-free c fan-out)
#define LOG2E     1.4426950408889634f
#define HALF_LOG2E 0.72134752044448170368f

typedef __attribute__((ext_vector_type(2))) float v2f;
typedef __attribute__((ext_vector_type(8))) float v8f;

__global__ __launch_bounds__(256)
void rbf_final_kernel(const float* __restrict__ x_grid,   // (NB, NGRID, NCH)
                      const float* __restrict__ h_grid,   // (NB, NGRID, NBASIS, NCH)
                      const float* __restrict__ target_x, // (NB, NTARGET, NCH)
                      const float* __restrict__ sigma,    // (NBASIS, NCH)
                      const float* __restrict__ g_w,      // (1, NBASIS)
                      const float* __restrict__ g_b,      // (1,)
                      float* __restrict__ out)            // (NB, NTARGET, NCH)
{
    __shared__ float hs[NBASIS * NCH * HS_STRIDE];  // [(k*8+c)][g'] padded
    __shared__ float xs[GCHUNK * NCH];              // flat copy of x chunk

    const int tid  = threadIdx.x;        // 0..255
    const int lane = tid & 31;           // wave32 lane
    const int b    = blockIdx.x >> 5;    // 32 blocks per batch
    const int tile = blockIdx.x & 31;    // 32 targets per block
    const int c    = tid & 7;
    const int t    = tile * 32 + (tid >> 3);

    const float y = target_x[(b * NTARGET + t) * NCH + c];

    // per-thread exp2 multipliers: -0.5*log2(e)/scale^2, scale = exp(sigma)+1e-6
    float m2[NBASIS];
#pragma unroll
    for (int k = 0; k < NBASIS; ++k) {
        float s = __builtin_amdgcn_exp2f(sigma[k * NCH + c] * LOG2E) + 1e-6f;
        m2[k] = -HALF_LOG2E / (s * s);
    }

    float acc0 = 0.f, acc1 = 0.f, acc2 = 0.f, acc3 = 0.f, acc4 = 0.f;

    for (int g0 = 0; g0 < NGRID; g0 += GCHUNK) {
        const float* hsrc = h_grid + ((size_t)(b * NGRID + g0)) * (NBASIS * NCH);
        const float* xsrc = x_grid + ((size_t)(b * NGRID + g0)) * NCH;

        // Stage h chunk: flat f = g'*40 + (k*8+c)  ->  hs[(k*8+c)*HS_STRIDE + g']
#pragma unroll
        for (int i = 0; i < (GCHUNK * NBASIS * NCH) / 256; ++i) {  // 20 loads/thread
            int f  = tid + i * 256;
            int gp = f / (NBASIS * NCH);
            int kc = f - gp * (NBASIS * NCH);
            hs[kc * HS_STRIDE + gp] = hsrc[f];
        }
        // Stage x chunk (already [g][c] layout -> flat copy)
#pragma unroll
        for (int i = 0; i < (GCHUNK * NCH) / 256; ++i) {           // 4 loads/thread
            int f = tid + i * 256;
            xs[f] = xsrc[f];
        }
        // Prefetch next chunk into cache while we compute this one
        if (g0 + GCHUNK < NGRID) {
            __builtin_prefetch(hsrc + GCHUNK * NBASIS * NCH + tid * 4, 0, 0);
        }
        __syncthreads();

#pragma unroll 4
        for (int gp = 0; gp < GCHUNK; ++gp) {
            float d  = xs[gp * NCH + c] - y;
            float d2 = d * d;
            acc0 += hs[(0 * NCH + c) * HS_STRIDE + gp] * __builtin_amdgcn_exp2f(d2 * m2[0]);
            acc1 += hs[(1 * NCH + c) * HS_STRIDE + gp] * __builtin_amdgcn_exp2f(d2 * m2[1]);
            acc2 += hs[(2 * NCH + c) * HS_STRIDE + gp] * __builtin_amdgcn_exp2f(d2 * m2[2]);
            acc3 += hs[(3 * NCH + c) * HS_STRIDE + gp] * __builtin_amdgcn_exp2f(d2 * m2[3]);
            acc4 += hs[(4 * NCH + c) * HS_STRIDE + gp] * __builtin_amdgcn_exp2f(d2 * m2[4]);
        }
        __syncthreads();
    }

    // ---- Final linear layer (nbasis -> 1) via V_WMMA_F32_16X16X4_F32 ----
    // A (16x4, documented layout: lane L -> M=L%16; VGPR0 = K0|K2, VGPR1 = K1|K3):
    //   A[m,k] = g_w[k] broadcast over rows.
    // B (4x16): columns = per-lane accumulator vectors, gathered with wave32 shuffles.
    // C seeded with bias; two K=4 WMMAs chain to cover nbasis=5 (zero-padded).
    const float gb = g_b[0];
    const float w0 = g_w[0], w1 = g_w[1], w2 = g_w[2], w3 = g_w[3], w4 = g_w[4];
    const bool lo = (lane < 16);

    v2f a_main; a_main.x = lo ? w0 : w2; a_main.y = lo ? w1 : w3;
    v2f a_tail; a_tail.x = lo ? w4 : 0.f; a_tail.y = 0.f;

    v8f cbias;
#pragma unroll
    for (int i = 0; i < 8; ++i) cbias[i] = gb;

    // WMMA #1: columns = accs of lanes 0..15
    {
        const int src = lane & 15;
        float s0 = __shfl(acc0, src), s1 = __shfl(acc1, src);
        float s2 = __shfl(acc2, src), s3 = __shfl(acc3, src);
        float s4 = __shfl(acc4, src);
        v2f bm; bm.x = lo ? s0 : s2; bm.y = lo ? s1 : s3;
        v2f bt; bt.x = lo ? s4 : 0.f; bt.y = 0.f;

        v8f d1 = __builtin_amdgcn_wmma_f32_16x16x4_f32(
            false, a_main, false, bm, (short)0, cbias, false, false);
        d1 = __builtin_amdgcn_wmma_f32_16x16x4_f32(
            false, a_tail, false, bt, (short)0, d1, false, false);

        // WMMA #2: columns = accs of lanes 16..31
        const int src2 = (lane & 15) + 16;
        float u0 = __shfl(acc0, src2), u1 = __shfl(acc1, src2);
        float u2 = __shfl(acc2, src2), u3 = __shfl(acc3, src2);
        float u4 = __shfl(acc4, src2);
        v2f bm2; bm2.x = lo ? u0 : u2; bm2.y = lo ? u1 : u3;
        v2f bt2; bt2.x = lo ? u4 : 0.f; bt2.y = 0.f;

        v8f d2 = __builtin_amdgcn_wmma_f32_16x16x4_f32(
            false, a_main, false, bm2, (short)0, cbias, false, false);
        d2 = __builtin_amdgcn_wmma_f32_16x16x4_f32(
            false, a_tail, false, bt2, (short)0, d2, false, false);

        // Lane L<16 result lives at D[M=0, N=L] (its VGPR0 of chain 1);
        // lane L>=16 result at D[M=8, N=L-16] (its VGPR0 of chain 2).
        float res = lo ? d1[0] : d2[0];
        out[blockIdx.x * 256 + tid] = res;
    }
}

extern "C" void kernel_launch(void* const* d_in, const int* in_sizes, int n_in,
                              void* d_out, int out_size, void* d_ws, size_t ws_size,
                              hipStream_t stream) {
    (void)in_sizes; (void)n_in; (void)out_size; (void)d_ws; (void)ws_size;
    const float* x_grid   = (const float*)d_in[0];
    const float* h_grid   = (const float*)d_in[1];
    const float* target_x = (const float*)d_in[2];
    const float* sigma    = (const float*)d_in[3];
    const float* g_w      = (const float*)d_in[4];
    const float* g_b      = (const float*)d_in[5];
    float* out = (float*)d_out;

    // 4 batches * 32 tiles = 128 blocks; 256 threads = one (t,c) each; exact cover.
    rbf_final_kernel<<<dim3(NB * 32), dim3(256), 0, stream>>>(
        x_grid, h_grid, target_x, sigma, g_w, g_b, out);
}